// NPDCLoss_63496796504335
// MI455X (gfx1250) — compile-verified
//
#include <hip/hip_runtime.h>

typedef float f4  __attribute__((ext_vector_type(4)));
typedef float v2f __attribute__((ext_vector_type(2)));
typedef float v8f __attribute__((ext_vector_type(8)));

#define N_LAG 50
#define NS    16384
#define D     2048
#define TPB   64                         // 2 waves per block
#define ROWS_PER_CHUNK 512
#define NCHUNKS (NS / ROWS_PER_CHUNK)    // 32
#define COL_TILES (D / (TPB * 4))        // 8 (each thread owns 4 columns)

// Pass 1: per-(chunk, column) partial sum of squared error, using a sliding
// 50-row window sum kept per-thread, with the 50-row lag history in LDS.
__global__ __launch_bounds__(TPB) void npdc_partial(const float* __restrict__ x_pred,
                                                    const float* __restrict__ x_true,
                                                    float* __restrict__ partial) {
    __shared__ f4 ring[N_LAG * TPB];     // 50 * 64 * 16B = 50 KB
    const int tid  = threadIdx.x;
    const int colv = blockIdx.x * TPB + tid;   // float4-column index
    const int t0   = blockIdx.y * ROWS_PER_CHUNK;
    const int dv   = D / 4;                    // 512 float4 per row

    const f4* xt4 = (const f4*)x_true;
    const f4* xp4 = (const f4*)x_pred;

    // Prologue: build window sum over rows [t0 - m0, t0), seed the ring.
    f4 S = {0.f, 0.f, 0.f, 0.f};
    const int m0 = (t0 < N_LAG) ? t0 : N_LAG;  // block-uniform
    for (int j = 1; j <= m0; ++j) {
        const int r = t0 - j;
        f4 v = xt4[r * dv + colv];
        S += v;
        ring[(r % N_LAG) * TPB + tid] = v;
    }

    f4 acc = {0.f, 0.f, 0.f, 0.f};
    int slot = t0 % N_LAG;
    const float inv_n = 1.0f / (float)N_LAG;

    for (int t = t0; t < t0 + ROWS_PER_CHUNK; ++t) {
        // Streaming, read-once data: non-temporal so L2 isn't thrashed.
        f4 xt = __builtin_nontemporal_load(&xt4[t * dv + colv]);
        f4 xp = __builtin_nontemporal_load(&xp4[t * dv + colv]);

        const int   m    = (t < N_LAG) ? t : N_LAG;      // block-uniform
        const float coef = (float)(N_LAG - m);           // 0 for t >= 50
        f4 avg  = (S + coef * xt) * inv_n;
        f4 diff = xp - avg;
        acc += diff * diff;

        if (t >= N_LAG) {                                // uniform branch
            f4 lag = ring[slot * TPB + tid];             // row t-50
            S += xt - lag;
        } else {
            S += xt;
        }
        ring[slot * TPB + tid] = xt;                     // row t replaces t-50
        slot = (slot + 1 == N_LAG) ? 0 : slot + 1;
    }

    ((f4*)partial)[blockIdx.y * dv + colv] = acc;        // [NCHUNKS][D]
}

// Pass 2: deterministic reduction of partial[NCHUNKS][D] over rows via WMMA:
// D = ones(16x4) x B(4x16) + C accumulated over k-chunks. With A == ones the
// column sums are invariant to the B K-slot layout; only N = lane%16 matters.
__global__ __launch_bounds__(32) void npdc_reduce(const float* __restrict__ partial,
                                                  float* __restrict__ out) {
    const int lane = threadIdx.x;                 // one wave32 per 16 columns
    const int c0   = blockIdx.x * 16;
    const int ncol = c0 + (lane & 15);
    const int rsel = (lane >> 4) << 1;            // lanes 0-15 -> rows k,k+1; 16-31 -> k+2,k+3

    v2f a; a.x = 1.0f; a.y = 1.0f;                // A = 16x4 all-ones
    v8f c = {0.f, 0.f, 0.f, 0.f, 0.f, 0.f, 0.f, 0.f};

    for (int kc = 0; kc < NCHUNKS; kc += 4) {     // 8 x v_wmma_f32_16x16x4_f32
        const int r = kc + rsel;
        v2f b;
        b.x = partial[r * D + ncol];
        b.y = partial[(r + 1) * D + ncol];
        c = __builtin_amdgcn_wmma_f32_16x16x4_f32(
                /*neg_a=*/false, a, /*neg_b=*/false, b,
                /*c_mod=*/(short)0, c, /*reuse_a=*/false, /*reuse_b=*/false);
    }

    if (lane < 16) {                              // every D row identical; row 0 in c[0]
        out[c0 + lane] = c[0] * (1.0f / (float)NS);
    }
}

extern "C" void kernel_launch(void* const* d_in, const int* in_sizes, int n_in,
                              void* d_out, int out_size, void* d_ws, size_t ws_size,
                              hipStream_t stream) {
    const float* x_pred = (const float*)d_in[0];
    const float* x_true = (const float*)d_in[1];
    float* out     = (float*)d_out;
    float* partial = (float*)d_ws;                // NCHUNKS * D * 4B = 256 KB scratch

    dim3 g1(COL_TILES, NCHUNKS);                  // 8 x 32 = 256 blocks, 512 waves
    npdc_partial<<<g1, TPB, 0, stream>>>(x_pred, x_true, partial);

    npdc_reduce<<<D / 16, 32, 0, stream>>>(partial, out);
}